// LogicConv2d_22351009808975
// MI455X (gfx1250) — compile-verified
//
#include <hip/hip_runtime.h>
#include <stdint.h>

// ---------------- problem constants (from reference setup_inputs) ----------
#define CC 32          // channels
#define HH 32          // input height
#define WW 32          // input width
#define KK 64          // gate networks
#define SS 64          // tree leaves (2^6)
#define NNODE 127      // total tree nodes per k (64+32+16+8+4+2+1)
#define OH 28
#define OW 28
#define BANDR 7        // output rows per band (4 bands * 7 = 28)
#define XROWS 11       // input rows needed per band: BANDR + 5 - 1
#define CSTRIDE (XROWS * WW)          // 352 floats per channel in LDS
#define XELEMS (CC * CSTRIDE)         // 11264 floats = 45056 B in LDS
#define COPY_UNITS (XELEMS / 4)       // 2816 x 16B units
#define UNITS_PER_C (XROWS * WW / 4)  // 88

// flat node offset of level l in the coef array: 0,64,96,112,120,124,126
#define NODEOFF(l) (128 - (128 >> (l)))

// ---------------- CDNA5 async global->LDS copy --------------------------
#if defined(__has_builtin)
# if __has_builtin(__builtin_amdgcn_global_load_async_to_lds_b128)
#  define HAVE_ASYNC_LDS 1
# endif
#endif
#ifndef HAVE_ASYNC_LDS
# define HAVE_ASYNC_LDS 0
#endif

typedef int v4i __attribute__((ext_vector_type(4)));
typedef __attribute__((address_space(1))) v4i as1_v4i;  // global 128-bit unit
typedef __attribute__((address_space(3))) v4i as3_v4i;  // LDS 128-bit unit

__device__ __forceinline__ as1_v4i* to_as1_b128(const void* p) {
  // generic->global: numeric value is the full 64-bit VA
  return (as1_v4i*)(uintptr_t)p;
}
__device__ __forceinline__ as3_v4i* to_as3_b128(void* p) {
  // generic LDS pointer: low 32 bits are the LDS byte address (ISA 10.2)
  return (as3_v4i*)(uint32_t)(uintptr_t)p;
}

__device__ __forceinline__ void wait_async0() {
  asm volatile("s_wait_asynccnt 0" ::: "memory");
}

// ---------------- kernel 1: softmax coefficients + gather bases ------------
// coef[k][node] = (c0, ca, cb, cab) of the 16-op softmax mixture.
// base[k][s]    = (baseA, baseB) LDS float-offsets (band-row-relative).
__global__ __launch_bounds__(256) void logic_prep(
    const int* __restrict__ idx_h, const int* __restrict__ idx_w,
    const int* __restrict__ idx_c,
    const float* __restrict__ w0, const float* __restrict__ w1,
    const float* __restrict__ w2, const float* __restrict__ w3,
    const float* __restrict__ w4, const float* __restrict__ w5,
    const float* __restrict__ w6,
    float4* __restrict__ coef, int2* __restrict__ base) {
  const int k = blockIdx.x;   // 64 blocks
  const int t = threadIdx.x;  // 256 threads

  if (t < NNODE) {
    const int flat = t;
    int lvl, n;
    if      (flat < 64)  { lvl = 0; n = flat; }
    else if (flat < 96)  { lvl = 1; n = flat - 64; }
    else if (flat < 112) { lvl = 2; n = flat - 96; }
    else if (flat < 120) { lvl = 3; n = flat - 112; }
    else if (flat < 124) { lvl = 4; n = flat - 120; }
    else if (flat < 126) { lvl = 5; n = flat - 124; }
    else                 { lvl = 6; n = 0; }
    const float* wl[7] = {w0, w1, w2, w3, w4, w5, w6};
    const float* wp = wl[lvl] + ((size_t)n * KK + k) * 16;

    float l[16];
    float m = -3.0e38f;
    #pragma unroll
    for (int j = 0; j < 16; ++j) { l[j] = wp[j]; m = fmaxf(m, l[j]); }
    float p[16], sum = 0.0f;
    #pragma unroll
    for (int j = 0; j < 16; ++j) { p[j] = __expf(l[j] - m); sum += p[j]; }
    const float inv = 1.0f / sum;
    #pragma unroll
    for (int j = 0; j < 16; ++j) p[j] *= inv;

    const float c0  = p[8]+p[9]+p[10]+p[11]+p[12]+p[13]+p[14]+p[15];
    const float ca  = p[2]+p[3]+p[6]+p[7]-p[8]-p[9]-p[12]-p[13];
    const float cb  = p[4]+p[5]+p[6]+p[7]-p[8]-p[9]-p[10]-p[11];
    const float cab = p[1]-p[2]-p[4]-2.0f*p[6]-p[7]+p[8]+2.0f*p[9]
                      +p[11]+p[13]-p[14];
    coef[k * NNODE + flat] = make_float4(c0, ca, cb, cab);
  } else if (t >= 128 && t < 128 + SS) {
    const int s = t - 128;
    const int ia = (k * 2 + 0) * SS + s;
    const int ib = (k * 2 + 1) * SS + s;
    base[k * SS + s] = make_int2(
        idx_c[ia] * CSTRIDE + idx_h[ia] * WW + idx_w[ia],
        idx_c[ib] * CSTRIDE + idx_h[ib] * WW + idx_w[ib]);
  }
}

// ---------------- kernel 2: main gather + gate-tree kernel -----------------
// blockIdx.x = band + 4*(kg + 8*b) ; band in [0,4), kg in [0,8), b in [0,B)
__global__ __launch_bounds__(256) void logic_main(
    const float* __restrict__ x, const float4* __restrict__ coef,
    const int2* __restrict__ base, float* __restrict__ out) {
  const int band = blockIdx.x & 3;
  const int kg   = (blockIdx.x >> 2) & 7;
  const int b    = blockIdx.x >> 5;
  const int r0   = band * BANDR;

  __shared__ float sX[XELEMS];  // 45056 B: 32 channels x 11 rows x 32 cols

  // --- stage x[b][:, r0:r0+11, :] into LDS via async copy ---
  const float* gx = x + (size_t)b * (CC * HH * WW) + r0 * WW;
  for (int u = threadIdx.x; u < COPY_UNITS; u += 256) {
    const int c = u / UNITS_PER_C;
    const int r = u - c * UNITS_PER_C;
    const float* src = gx + c * (HH * WW) + r * 4;
    float* dst = &sX[c * CSTRIDE + r * 4];
#if HAVE_ASYNC_LDS
    __builtin_amdgcn_global_load_async_to_lds_b128(
        to_as1_b128(src), to_as3_b128(dst), /*offset=*/0, /*cpol=*/0);
#else
    *(float4*)dst = *(const float4*)src;
#endif
  }
  wait_async0();
  __syncthreads();

  // --- one output position per lane (196 active of 256) ---
  const int tid = (int)threadIdx.x;
  const bool active = tid < BANDR * OW;
  const int p  = active ? tid : 0;
  const int oi = p / OW;
  const int oj = p - oi * OW;
  const int off = oi * WW + oj;  // relative LDS offset within a channel block

  for (int kk = 0; kk < 8; ++kk) {
    const int k = kg * 8 + kk;
    const float4* __restrict__ ck = coef + k * NNODE;  // uniform addresses
    const int2*   __restrict__ bk = base + k * SS;     // uniform addresses

    float stk[6];
    float v = 0.0f;
    #pragma unroll
    for (int s = 0; s < SS; ++s) {
      const int2  bs = bk[s];          // scalar (wave-uniform) load
      const float a  = sX[bs.x + off]; // ds_load_b32, stride-1 across lanes
      const float bb = sX[bs.y + off];
      const float4 c = ck[s];          // scalar (wave-uniform) b128 load
      v = fmaf(c.w, a * bb, fmaf(c.z, bb, fmaf(c.y, a, c.x)));
      // depth-first merge of completed subtrees (trailing 1-bits of s)
      #pragma unroll
      for (int lvl = 0; lvl < 6; ++lvl) {
        if (((s >> lvl) & 1) == 0) { stk[lvl] = v; break; }
        const int n = s >> (lvl + 1);
        const float4 cc = ck[NODEOFF(lvl + 1) + n];
        const float aa = stk[lvl];  // even child = 'a', v = odd child = 'b'
        v = fmaf(cc.w, aa * v, fmaf(cc.z, v, fmaf(cc.y, aa, cc.x)));
      }
    }
    if (active) {
      const int orow = r0 + oi;
      out[(((size_t)b * KK + k) * OH + orow) * OW + oj] = v;
    }
  }
}

// ---------------- launcher -------------------------------------------------
extern "C" void kernel_launch(void* const* d_in, const int* in_sizes, int n_in,
                              void* d_out, int out_size, void* d_ws,
                              size_t ws_size, hipStream_t stream) {
  const float* x     = (const float*)d_in[0];
  const int*   idx_h = (const int*)d_in[1];
  const int*   idx_w = (const int*)d_in[2];
  const int*   idx_c = (const int*)d_in[3];
  const float* w0 = (const float*)d_in[4];
  const float* w1 = (const float*)d_in[5];
  const float* w2 = (const float*)d_in[6];
  const float* w3 = (const float*)d_in[7];
  const float* w4 = (const float*)d_in[8];
  const float* w5 = (const float*)d_in[9];
  const float* w6 = (const float*)d_in[10];

  float4* coef = (float4*)d_ws;                                   // 130048 B
  int2*   base = (int2*)((char*)d_ws + (size_t)KK * NNODE * 16);  // +32768 B

  const int B = in_sizes[0] / (CC * HH * WW);  // 16

  logic_prep<<<KK, 256, 0, stream>>>(idx_h, idx_w, idx_c,
                                     w0, w1, w2, w3, w4, w5, w6, coef, base);
  logic_main<<<B * 8 * 4, 256, 0, stream>>>(x, coef, base, (float*)d_out);
}